// LandmarkImageLayer_90168543412831
// MI455X (gfx1250) — compile-verified
//
#include <hip/hip_runtime.h>

#define HALFW 16
#define PATCH 33
#define KTOT  (PATCH * PATCH)   // 1089
#define IMH   512
#define IMW   512

typedef __attribute__((ext_vector_type(2))) float v2f;
typedef __attribute__((ext_vector_type(8))) float v8f;

// ---------------- zero-init: wide b128 stores ----------------
__global__ void zero_f4_kernel(float4* __restrict__ out, size_t n4) {
    size_t i = (size_t)blockIdx.x * blockDim.x + threadIdx.x;
    if (i < n4) out[i] = make_float4(0.f, 0.f, 0.f, 0.f);
}

// ---------------- WMMA splat ----------------
// One wave per (image b, k-tile, l-tile). Squared distance tile computed as
//   D[k,l] = A(16x4) * B(4x16) with A_k=[o0,o1,|o|^2,1], B_l=[-2d0,-2d1,1,|d|^2]^T
// via v_wmma_f32_16x16x4_f32, then val = 1/(1+sqrt(D+1e-6)) scattered with
// no-return global_atomic_add_f32. sqrt/rcp use the ~1ulp hardware approx ops
// (argument ranges are benign: sqrt in [1e-6, ~520], rcp in [1, ~24]).
__global__ __launch_bounds__(128) void splat_wmma_kernel(
        const float* __restrict__ lms, float* __restrict__ out,
        int B, int L, int ktiles, int ltiles) {
    const int lane = threadIdx.x & 31;
    const int wid  = (int)((blockIdx.x * blockDim.x + threadIdx.x) >> 5);
    const int wavesPerImg = ktiles * ltiles;
    const int total = B * wavesPerImg;
    if (wid >= total) return;              // wave-uniform exit (EXEC stays all-1s)

    const int b  = wid / wavesPerImg;
    const int r0 = wid - b * wavesPerImg;
    const int kt = r0 / ltiles;
    const int lt = r0 - kt * ltiles;

    const int m  = lane & 15;
    const int hi = lane >> 4;

    // ---- A matrix: offset rows, augmented to inner-dim 4 ----
    int ka = kt * 16 + m;
    if (ka >= KTOT) ka = KTOT - 1;         // padded rows, masked at scatter
    const float o0 = (float)(ka % PATCH - HALFW);
    const float o1 = (float)(ka / PATCH - HALFW);
    v2f a;
    if (!hi) { a.x = o0;               a.y = o1;   }   // K=0, K=1
    else     { a.x = o0*o0 + o1*o1;    a.y = 1.0f; }   // K=2, K=3

    // ---- B matrix: landmark columns ----
    const int l  = lt * 16 + m;
    const int lc = (l < L) ? l : (L - 1);
    const float* lp = lms + ((size_t)b * L + lc) * 2;
    float lm0 = lp[0], lm1 = lp[1];
    const float clo = (float)HALFW, chi = (float)(IMH - 1 - HALFW);
    lm0 = fminf(fmaxf(lm0, clo), chi);
    lm1 = fminf(fmaxf(lm1, clo), chi);
    const float f0 = floorf(lm0), f1 = floorf(lm1);
    const float d0 = lm0 - f0,    d1 = lm1 - f1;
    v2f bv;
    if (!hi) { bv.x = -2.0f * d0;  bv.y = -2.0f * d1;    }  // K=0, K=1
    else     { bv.x = 1.0f;        bv.y = d0*d0 + d1*d1; }  // K=2, K=3

    // ---- 16x16x4 f32 WMMA: D = A*B (+0) ----
    v8f c = {};
    c = __builtin_amdgcn_wmma_f32_16x16x4_f32(
            false, a, false, bv, (short)0, c, false, false);

    // ---- scatter: lane holds D[M = r + 8*hi][N = m] for r=0..7 ----
    // patch base pointer: landmark top-left corner of the 33x33 patch
    float* pbase = out + (size_t)b * (IMH * IMW)
                 + ((int)f0 + IMW * (int)f1) - HALFW - HALFW * IMW;
    const bool lvalid = (l < L);
    const int  k0   = kt * 16 + 8 * hi;    // first k this lane scatters
    int col = k0 % PATCH;                  // fast axis within patch [0,33)
    int row = k0 / PATCH;                  // slow axis within patch
    #pragma unroll
    for (int r = 0; r < 8; ++r) {
        const float s   = __builtin_amdgcn_sqrtf(c[r] + 1e-6f);
        const float val = __builtin_amdgcn_rcpf(1.0f + s);
        if (lvalid && (k0 + r) < KTOT) {
            float* p = pbase + (col + row * IMW);
            __hip_atomic_fetch_add(p, val, __ATOMIC_RELAXED,
                                   __HIP_MEMORY_SCOPE_AGENT);
        }
        // incremental k -> (row, col) walk, no div/mod in the loop
        if (++col == PATCH) { col = 0; ++row; }
    }
}

extern "C" void kernel_launch(void* const* d_in, const int* in_sizes, int n_in,
                              void* d_out, int out_size, void* d_ws, size_t ws_size,
                              hipStream_t stream) {
    (void)d_ws; (void)ws_size; (void)n_in;
    // d_in[0] = img (unused by the math), d_in[1] = landmarks [B,L,2] f32
    const float* lms = (const float*)d_in[1];
    float* out = (float*)d_out;

    const int B = in_sizes[0] / (IMH * IMW);      // 64
    const int L = in_sizes[1] / (2 * B);          // 106

    // 1) zero-init output (64 MB) with b128 stores
    const size_t n4 = (size_t)out_size / 4;       // out_size divisible by 4
    const int zb = (int)((n4 + 255) / 256);
    zero_f4_kernel<<<zb, 256, 0, stream>>>((float4*)out, n4);

    // 2) WMMA splat
    const int ktiles = (KTOT + 15) / 16;          // 69
    const int ltiles = (L + 15) / 16;             // 7
    const long long waves = (long long)B * ktiles * ltiles;
    const int blocks = (int)((waves * 32 + 127) / 128);
    splat_wmma_kernel<<<blocks, 128, 0, stream>>>(lms, out, B, L, ktiles, ltiles);
}